// DM_30133490549587
// MI455X (gfx1250) — compile-verified
//
#include <hip/hip_runtime.h>

typedef __attribute__((ext_vector_type(2))) float v2f;
typedef __attribute__((ext_vector_type(4))) float v4f;
typedef __attribute__((ext_vector_type(8))) float v8f;

#define N_WORDS 100000
#define EMB     128
#define CTX     8
#define KOUT    6

// ---------------------------------------------------------------------------
// Kernel 1: tiled transpose Wp [EMB, N_WORDS] -> WpT [N_WORDS, EMB]
// N_WORDS = 3125*32 and EMB = 4*32, so no edge handling needed.
// ---------------------------------------------------------------------------
__global__ void DM_transpose_wp(const float* __restrict__ Wp,
                                float* __restrict__ WpT) {
    __shared__ float tile[32][33];              // +1 pad: no LDS bank conflicts
    const int tx = threadIdx.x;                 // 0..31
    const int ty = threadIdx.y;                 // 0..7
    const int bx = blockIdx.x;                  // word tile (0..3124)
    const int by = blockIdx.y;                  // emb tile  (0..3)

#pragma unroll
    for (int j = 0; j < 4; ++j) {
        const int d = by * 32 + ty + j * 8;     // emb index (row of Wp)
        const int w = bx * 32 + tx;             // word index (contiguous)
        tile[ty + j * 8][tx] = Wp[(size_t)d * N_WORDS + w];
    }
    __syncthreads();
#pragma unroll
    for (int j = 0; j < 4; ++j) {
        const int w = bx * 32 + ty + j * 8;     // word index
        const int d = by * 32 + tx;             // emb index (contiguous)
        WpT[(size_t)w * EMB + d] = tile[tx][ty + j * 8];
    }
}

// ---------------------------------------------------------------------------
// Kernel 2: one wave (32 lanes) per batch element b.
//   Phase A: x[b] = D[doc_b] + sum_c W[ctx_{b,c}]  -> LDS (128 floats)
//   Phase B: out[b, m] = x[b] . WpT[id_{b,m}]  via V_WMMA_F32_16X16X4_F32
//     A[m,k] = WpT[id_m, d0+k]   (rows m>=KOUT carry duplicated data; the
//                                 matmul keeps rows independent, and C rows
//                                 m>=KOUT are never stored, so no masking
//                                 or exec manipulation is needed)
//     B[k,n] = x[d0+k]           (broadcast across N)
//     C[m,0] accumulates the dot -> lane 0, c[m]
// ---------------------------------------------------------------------------
template <bool USE_TRANSPOSED>
__global__ void DM_nce_dot(const float* __restrict__ D,
                           const float* __restrict__ W,
                           const float* __restrict__ WpOrT,
                           const int*   __restrict__ ctx_ids,
                           const int*   __restrict__ doc_ids,
                           const int*   __restrict__ tn_ids,
                           float*       __restrict__ out) {
    __shared__ float xs[EMB];
    const int b    = blockIdx.x;
    const int lane = threadIdx.x;               // 0..31 (wave32)

    // -------- Phase A: build x[b] in LDS (each lane owns 4 contiguous d's)
    const int doc = doc_ids[b];
    v4f acc = *(const v4f*)(D + (size_t)doc * EMB + lane * 4);
#pragma unroll
    for (int c = 0; c < CTX; ++c) {
        const int w = ctx_ids[b * CTX + c];     // may be N_WORDS (zero pad row)
        acc += *(const v4f*)(W + (size_t)w * EMB + lane * 4);
    }
    *(v4f*)(xs + lane * 4) = acc;
    __syncthreads();

    // -------- Phase B: WMMA accumulation over 32 chunks of K=4
    const int m = lane & 15;                    // A-matrix row / output index
    const int g = lane >> 4;                    // lane-group: K pair {2g,2g+1}
    // Clamped, branch-free id gather: lanes m>=KOUT duplicate id[KOUT-1].
    // Their loads merge with m==KOUT-1 lanes' requests in the coalescer and
    // their C rows are discarded.
    const int mc = (m < KOUT) ? m : (KOUT - 1);
    const int id = tn_ids[b * KOUT + mc];

    v8f cacc = {};
#pragma unroll 8
    for (int d0 = 0; d0 < EMB; d0 += 4) {
        const int d = d0 + 2 * g;
        // B operand: x[d], x[d+1] broadcast from LDS
        v2f bvec;
        bvec.x = xs[d];
        bvec.y = xs[d + 1];
        // A operand: WpT[id_m, d], WpT[id_m, d+1] -- unconditional load
        v2f avec;
        if (USE_TRANSPOSED) {
            avec = *(const v2f*)(WpOrT + (size_t)id * EMB + d);
        } else {
            avec.x = WpOrT[(size_t)d * N_WORDS + id];
            avec.y = WpOrT[(size_t)(d + 1) * N_WORDS + id];
        }
        // (neg_a, A, neg_b, B, c_mod, C, reuse_a, reuse_b)
        cacc = __builtin_amdgcn_wmma_f32_16x16x4_f32(
            false, avec, false, bvec, (short)0, cacc, false, false);
    }

    // C[m, n=0] lives in lane 0, VGPR m (m < 8)
    if (lane == 0) {
        float* o = out + (size_t)b * KOUT;
        o[0] = cacc[0];
        o[1] = cacc[1];
        o[2] = cacc[2];
        o[3] = cacc[3];
        o[4] = cacc[4];
        o[5] = cacc[5];
    }
}

// ---------------------------------------------------------------------------
extern "C" void kernel_launch(void* const* d_in, const int* in_sizes, int n_in,
                              void* d_out, int out_size, void* d_ws, size_t ws_size,
                              hipStream_t stream) {
    const float* D   = (const float*)d_in[0];
    const float* W   = (const float*)d_in[1];
    const float* Wp  = (const float*)d_in[2];
    const int*   ctx = (const int*)d_in[3];
    const int*   doc = (const int*)d_in[4];
    const int*   tn  = (const int*)d_in[5];
    float*       out = (float*)d_out;

    const int B = in_sizes[4];                  // doc_ids element count = 16384

    const size_t wpt_bytes = (size_t)N_WORDS * EMB * sizeof(float); // 51.2 MB
    if (ws_size >= wpt_bytes) {
        float* WpT = (float*)d_ws;
        dim3 tb(32, 8);
        dim3 tg(N_WORDS / 32, EMB / 32);        // 3125 x 4, exact
        DM_transpose_wp<<<tg, tb, 0, stream>>>(Wp, WpT);
        DM_nce_dot<true><<<B, 32, 0, stream>>>(D, W, WpT, ctx, doc, tn, out);
    } else {
        // Fallback: strided column reads straight out of Wp (L2-resident).
        DM_nce_dot<false><<<B, 32, 0, stream>>>(D, W, Wp, ctx, doc, tn, out);
    }
}